// SynapticSpike_14396730376458
// MI455X (gfx1250) — compile-verified
//
#include <hip/hip_runtime.h>
#include <stdint.h>

// ---------------------------------------------------------------------------
// RSynaptic single-step, zero initial state:
//   rec_conv(0) == bias broadcast          -> syn = x + conv_b[t]
//   reset = (0 - thr > 0) = (thr < 0)      -> mem = syn - (thr<0 ? thr : 0)
//   spk  = (mem - thr > 0) ? 1 : 0
// Pure streaming: ~470 MB traffic -> ~20 us at 23.3 TB/s. No matrix math
// survives, so the CDNA5-relevant path is data movement: B128 NT streams +
// async global->LDS double-buffered pipeline (ASYNCcnt).
// ---------------------------------------------------------------------------

typedef float v4f __attribute__((ext_vector_type(4)));
typedef int   v4i __attribute__((ext_vector_type(4)));

static constexpr uint32_t kT = 112;
static constexpr size_t   kTotal = (size_t)4 * 112 * 2048 * 32;   // 29360128
static constexpr uint32_t kVec = (uint32_t)(kTotal / 4);          // 7340032 vec4
static constexpr uint32_t kStageVec = 1024;                       // vec4 per block-stage
static constexpr uint32_t kStages = kVec / kStageVec;             // 7168 (exact)
static constexpr uint32_t kGrid = 1792;                           // 4 stages per block (exact)
static constexpr int      kUnroll = 4;                            // vec4 per lane per stage
static constexpr uint32_t kWaveStageBytes = 32u * kUnroll * 16u;  // 2048 B

#if defined(__AMDGCN__) && defined(__gfx1250__) && \
    __has_builtin(__builtin_amdgcn_global_load_async_to_lds_b128)
#define USE_ASYNC_LDS 1
#else
#define USE_ASYNC_LDS 0
#endif

#if USE_ASYNC_LDS
typedef __attribute__((address_space(1))) v4i gv4i;  // global int4
typedef __attribute__((address_space(3))) v4i lv4i;  // LDS int4

#if __has_builtin(__builtin_amdgcn_s_wait_asynccnt)
#define WAIT_ASYNC(N)                          \
  do {                                         \
    __builtin_amdgcn_s_wait_asynccnt(N);       \
    asm volatile("" ::: "memory");             \
  } while (0)
#else
#define WAIT_ASYNC(N) asm volatile("s_wait_asynccnt " #N ::: "memory")
#endif
#endif

__global__ __launch_bounds__(256) void rsyn_stream_kernel(
    const float* __restrict__ x, const float* __restrict__ conv_b,
    const float* __restrict__ thr_p, float* __restrict__ o_spk,
    float* __restrict__ o_mem, float* __restrict__ o_syn) {
  const uint32_t wave = threadIdx.x >> 5;
  const uint32_t lane = threadIdx.x & 31u;

  const float thr = thr_p[0];
  const float reset_amt = (thr < 0.0f) ? thr : 0.0f;  // reset*thr with mem0==0

  const v4f* __restrict__ xv4 = (const v4f*)x;
  v4f* __restrict__ sv4 = (v4f*)o_spk;
  v4f* __restrict__ mv4 = (v4f*)o_mem;
  v4f* __restrict__ yv4 = (v4f*)o_syn;

#if USE_ASYNC_LDS
  // 8 waves x 2 buffers x 2KB = 32KB LDS; each wave owns a private slice,
  // so the async pipeline needs no block barriers.
  __shared__ char smem[8 * 2 * kWaveStageBytes];
  const uint32_t lwbase = wave * (2u * kWaveStageBytes) + lane * 16u;

  auto issue = [&](uint32_t s, uint32_t buf) {
    const uint32_t vbase = s * kStageVec + wave * (32u * kUnroll) + lane;
    const uint32_t lbase = lwbase + buf * kWaveStageBytes;
#pragma unroll
    for (int u = 0; u < kUnroll; ++u) {
      const void* g = (const void*)(xv4 + (vbase + 32u * (uint32_t)u));
      void* l = (void*)(smem + lbase + 512u * (uint32_t)u);
      // flat LDS address low 32 bits == LDS byte offset (ISA 10.2 aperture map)
      __builtin_amdgcn_global_load_async_to_lds_b128(
          (gv4i*)(uintptr_t)g,
          (lv4i*)(uint32_t)(uintptr_t)l,
          /*offset=*/0, /*cpol=*/0);
    }
  };

  uint32_t buf = 0;
  issue(blockIdx.x, 0);
  for (uint32_t s = blockIdx.x; s < kStages; s += gridDim.x) {
    const uint32_t snext = s + gridDim.x;
    if (snext < kStages) {
      issue(snext, buf ^ 1u);   // prefetch next stage into other buffer
      WAIT_ASYNC(4);            // wait for current stage only (in-order)
    } else {
      WAIT_ASYNC(0);
    }
    const uint32_t vbase = s * kStageVec + wave * (32u * kUnroll) + lane;
    const uint32_t lbase = lwbase + buf * kWaveStageBytes;
    // stage never crosses a t-boundary (16384 vec4 per t, 16384 % 1024 == 0)
    const float bb = conv_b[(s >> 4) % kT];
#pragma unroll
    for (int u = 0; u < kUnroll; ++u) {
      const uint32_t v = vbase + 32u * (uint32_t)u;
      v4f xv = *(const v4f*)(smem + lbase + 512u * (uint32_t)u);
      v4f syn = xv + bb;
      v4f mem = syn - reset_amt;
      v4f spk;
      spk.x = ((mem.x - thr) > 0.0f) ? 1.0f : 0.0f;
      spk.y = ((mem.y - thr) > 0.0f) ? 1.0f : 0.0f;
      spk.z = ((mem.z - thr) > 0.0f) ? 1.0f : 0.0f;
      spk.w = ((mem.w - thr) > 0.0f) ? 1.0f : 0.0f;
      __builtin_nontemporal_store(spk, sv4 + v);
      __builtin_nontemporal_store(mem, mv4 + v);
      __builtin_nontemporal_store(syn, yv4 + v);
    }
    buf ^= 1u;
  }
#else
  // Fallback: direct non-temporal B128 streaming (no async builtins found).
  for (uint32_t s = blockIdx.x; s < kStages; s += gridDim.x) {
    const uint32_t vbase = s * kStageVec + wave * (32u * kUnroll) + lane;
    const float bb = conv_b[(s >> 4) % kT];
#pragma unroll
    for (int u = 0; u < kUnroll; ++u) {
      const uint32_t v = vbase + 32u * (uint32_t)u;
      v4f xv = __builtin_nontemporal_load(xv4 + v);
      v4f syn = xv + bb;
      v4f mem = syn - reset_amt;
      v4f spk;
      spk.x = ((mem.x - thr) > 0.0f) ? 1.0f : 0.0f;
      spk.y = ((mem.y - thr) > 0.0f) ? 1.0f : 0.0f;
      spk.z = ((mem.z - thr) > 0.0f) ? 1.0f : 0.0f;
      spk.w = ((mem.w - thr) > 0.0f) ? 1.0f : 0.0f;
      __builtin_nontemporal_store(spk, sv4 + v);
      __builtin_nontemporal_store(mem, mv4 + v);
      __builtin_nontemporal_store(syn, yv4 + v);
    }
  }
#endif
}

extern "C" void kernel_launch(void* const* d_in, const int* in_sizes, int n_in,
                              void* d_out, int out_size, void* d_ws,
                              size_t ws_size, hipStream_t stream) {
  // setup_inputs order: x, alpha, beta, conv_w, conv_b, threshold
  const float* x = (const float*)d_in[0];
  const float* conv_b = (const float*)d_in[4];
  const float* thr = (const float*)d_in[5];
  float* out = (float*)d_out;  // [spk | mem | syn], each kTotal floats
  (void)in_sizes; (void)n_in; (void)out_size; (void)d_ws; (void)ws_size;

  hipLaunchKernelGGL(rsyn_stream_kernel, dim3(kGrid), dim3(256), 0, stream,
                     x, conv_b, thr, out, out + kTotal, out + 2 * kTotal);
}